// CFMLoss_73512660238528
// MI455X (gfx1250) — compile-verified
//
#include <hip/hip_runtime.h>
#include <math.h>

#define NN 8192
#define DD 128
#define CSPLIT 4            // column splits (gridDim.y)
#define TILES_PER_SPLIT ((NN / 16) / CSPLIT)   // 128 column tiles per split
#define BSTRIDE 132         // padded LDS row stride (floats) -> conflict-free ds_load_b64

typedef __attribute__((ext_vector_type(2))) float v2f;
typedef __attribute__((ext_vector_type(8))) float v8f;

// ---------------------------------------------------------------------------
// Kernel 1: per-row precompute.  ws_sq2[i] = ||z2_i||^2
//           ws_diag[i] = a_ii = 2*<z1_i, z2_i> - ||z2_i||^2
// ---------------------------------------------------------------------------
__global__ __launch_bounds__(256) void cfm_prep(const float* __restrict__ z1,
                                                const float* __restrict__ z2,
                                                float* __restrict__ ws_sq2,
                                                float* __restrict__ ws_diag) {
  int i = blockIdx.x * 256 + threadIdx.x;
  const float4* p1 = (const float4*)(z1 + (size_t)i * DD);
  const float4* p2 = (const float4*)(z2 + (size_t)i * DD);
  float s2 = 0.f, dt = 0.f;
#pragma unroll
  for (int k = 0; k < DD / 4; ++k) {
    float4 a = p1[k], b = p2[k];
    s2 += b.x * b.x + b.y * b.y + b.z * b.z + b.w * b.w;
    dt += a.x * b.x + a.y * b.y + a.z * b.z + a.w * b.w;
  }
  ws_sq2[i]  = s2;
  ws_diag[i] = 2.f * dt - s2;
}

// ---------------------------------------------------------------------------
// Kernel 2: fused GEMM + streaming logsumexp, B staged via double-buffered LDS.
// Block = 128 threads = 4 waves; block owns 64 rows (16 per wave); all waves
// consume the SAME column tile. asm scheduling barriers force all 16 DS loads
// to issue before the 32-WMMA chain (incremental dscnt waits, latency hidden).
// ---------------------------------------------------------------------------
__global__ __launch_bounds__(128) void cfm_lse(const float* __restrict__ z1,
                                               const float* __restrict__ z2,
                                               const float* __restrict__ sq2,
                                               float* __restrict__ m_part,
                                               float* __restrict__ s_part) {
  const int tid  = threadIdx.x;
  const int wave = tid >> 5;          // 0..3 : row sub-tile
  const int lane = tid & 31;          // wave32
  const int hf   = lane >> 4;         // half-wave -> K pair select
  const int l16  = lane & 15;
  const int row_base = blockIdx.x * 64 + wave * 16;
  const int split    = blockIdx.y;
  const int ct0      = split * TILES_PER_SPLIT;

  __shared__ float bbuf[2][16 * BSTRIDE];   // double-buffered B tile (16 cols x 128 K)

  // Preload A = 2*z1 slice (ISA 16x4 f32 A layout: lanes 0-15 hold K0,K1;
  // lanes 16-31 hold K2,K3; M = lane&15). 32 x v2f = 64 VGPRs, loaded once.
  v2f a[DD / 4];
  {
    const float* ap = z1 + (size_t)(row_base + l16) * DD + hf * 2;
#pragma unroll
    for (int k = 0; k < DD / 4; ++k) a[k] = *(const v2f*)(ap + k * 4) * 2.0f;
  }

  float m[8], s[8];
#pragma unroll
  for (int r = 0; r < 8; ++r) { m[r] = -INFINITY; s[r] = 0.f; }

  // Cooperative staging geometry: thread t loads 16 consecutive floats of
  // z2 row (col_base + t>>3), k-offset (t&7)*16.   128 thr x 16 = 2048 floats.
  const int scol = tid >> 3;          // 0..15
  const int skof = (tid & 7) * 16;    // 0..112

  // Prologue: stage tile ct0 into buffer 0.
  {
    const float4* gp = (const float4*)(z2 + (size_t)(ct0 * 16 + scol) * DD + skof);
    float4 r0 = gp[0], r1 = gp[1], r2 = gp[2], r3 = gp[3];
    float* dp = &bbuf[0][scol * BSTRIDE + skof];
    *(float4*)(dp)      = r0;
    *(float4*)(dp + 4)  = r1;
    *(float4*)(dp + 8)  = r2;
    *(float4*)(dp + 12) = r3;
  }
  __syncthreads();

  for (int i = 0; i < TILES_PER_SPLIT; ++i) {
    const int cur = i & 1;
    const int col_base = (ct0 + i) * 16;

    // Issue next tile's global loads + this tile's sq2 up front.
    float4 r0, r1, r2, r3;
    const bool more = (i + 1) < TILES_PER_SPLIT;
    if (more) {
      const float4* gp =
          (const float4*)(z2 + (size_t)((ct0 + i + 1) * 16 + scol) * DD + skof);
      r0 = gp[0]; r1 = gp[1]; r2 = gp[2]; r3 = gp[3];
    }
    const float sc = sq2[col_base + l16];

    // Barrier #1: pin the global prefetch above the DS traffic.
    asm volatile("" ::: "memory");

    // Pull the whole B slice into registers (16 paired DS loads)...
    const float* bl = &bbuf[cur][l16 * BSTRIDE + hf * 2];
    v2f b[DD / 4];
#pragma unroll
    for (int k = 0; k < DD / 4; ++k) b[k] = *(const v2f*)(bl + 4 * k);

    // Barrier #2: all DS loads issued before any WMMA -> incremental dscnt
    // waits; only the first load's latency is exposed.
    asm volatile("" ::: "memory");

    v8f c = {};
#pragma unroll
    for (int k = 0; k < DD / 4; ++k)
      c = __builtin_amdgcn_wmma_f32_16x16x4_f32(
          /*neg_a=*/false, a[k], /*neg_b=*/false, b[k],
          /*c_mod=*/(short)0, c, /*reuse_a=*/false, /*reuse_b=*/false);

    // Online softmax, one exp per row:  d = x-m, e = exp(-|d|)
    //   d<=0: s += e, m unchanged;  d>0: s = s*e + 1, m = x.
#pragma unroll
    for (int r = 0; r < 8; ++r) {
      float x  = c[r] - sc;                 // a_ij = 2*d_ij - sq2_j
      float d  = x - m[r];
      float e  = __expf(-fabsf(d));
      float t0 = s[r] + e;
      float t1 = fmaf(s[r], e, 1.0f);
      bool  le = (d <= 0.0f);
      s[r] = le ? t0 : t1;
      m[r] = le ? m[r] : x;
    }

    // Store next tile into the other buffer; safe: previous barrier ensured
    // everyone finished reading it (it was 'cur' in the previous iteration).
    if (more) {
      float* dp = &bbuf[cur ^ 1][scol * BSTRIDE + skof];
      *(float4*)(dp)      = r0;
      *(float4*)(dp + 4)  = r1;
      *(float4*)(dp + 8)  = r2;
      *(float4*)(dp + 12) = r3;
    }
    __syncthreads();
  }

  // Butterfly combine across the 16 lanes of each half (columns), one exp each.
#pragma unroll
  for (int off = 1; off < 16; off <<= 1) {
#pragma unroll
    for (int r = 0; r < 8; ++r) {
      float m2  = __shfl_xor(m[r], off, 32);
      float s2v = __shfl_xor(s[r], off, 32);
      float d   = m2 - m[r];
      float e   = __expf(-fabsf(d));
      float t0  = fmaf(s2v, e, s[r]);   // m2 <= m
      float t1  = fmaf(s[r], e, s2v);   // m2 >  m
      bool  le  = (d <= 0.0f);
      s[r] = le ? t0 : t1;
      m[r] = le ? m[r] : m2;
    }
  }

  // lane 0 holds rows 0-7 combined; lane 16 holds rows 8-15.
  if (l16 == 0) {
    const int rb = row_base + hf * 8;
#pragma unroll
    for (int r = 0; r < 8; ++r) {
      m_part[(size_t)split * NN + rb + r] = m[r];
      s_part[(size_t)split * NN + rb + r] = s[r];
    }
  }
}

// ---------------------------------------------------------------------------
// Kernel 3: deterministic merge of column-split partials + final reduction.
// out = mean_i( (M_i + log S_i) - a_ii )
// ---------------------------------------------------------------------------
__global__ __launch_bounds__(256) void cfm_finish(const float* __restrict__ m_part,
                                                  const float* __restrict__ s_part,
                                                  const float* __restrict__ diag,
                                                  float* __restrict__ out) {
  __shared__ float sh[256];
  float acc = 0.f;
  for (int row = threadIdx.x; row < NN; row += 256) {
    float M = -INFINITY;
#pragma unroll
    for (int p = 0; p < CSPLIT; ++p) M = fmaxf(M, m_part[(size_t)p * NN + row]);
    float S = 0.f;
#pragma unroll
    for (int p = 0; p < CSPLIT; ++p)
      S += s_part[(size_t)p * NN + row] * __expf(m_part[(size_t)p * NN + row] - M);
    acc += (M + __logf(S)) - diag[row];
  }
  sh[threadIdx.x] = acc;
  __syncthreads();
#pragma unroll
  for (int st = 128; st > 0; st >>= 1) {
    if ((int)threadIdx.x < st) sh[threadIdx.x] += sh[threadIdx.x + st];
    __syncthreads();
  }
  if (threadIdx.x == 0) out[0] = sh[0] * (1.0f / (float)NN);
}

// ---------------------------------------------------------------------------
extern "C" void kernel_launch(void* const* d_in, const int* in_sizes, int n_in,
                              void* d_out, int out_size, void* d_ws, size_t ws_size,
                              hipStream_t stream) {
  (void)in_sizes; (void)n_in; (void)out_size; (void)ws_size;
  const float* z1 = (const float*)d_in[0];
  const float* z2 = (const float*)d_in[1];
  float* out = (float*)d_out;

  float* ws_sq2  = (float*)d_ws;            // [NN]
  float* ws_diag = ws_sq2 + NN;             // [NN]
  float* ws_m    = ws_diag + NN;            // [CSPLIT*NN]
  float* ws_s    = ws_m + (size_t)CSPLIT * NN; // [CSPLIT*NN]

  cfm_prep<<<NN / 256, 256, 0, stream>>>(z1, z2, ws_sq2, ws_diag);
  cfm_lse<<<dim3(NN / 64, CSPLIT, 1), 128, 0, stream>>>(z1, z2, ws_sq2, ws_m, ws_s);
  cfm_finish<<<1, 256, 0, stream>>>(ws_m, ws_s, ws_diag, out);
}